// ROIResize_60464549593192
// MI455X (gfx1250) — compile-verified
//
#include <hip/hip_runtime.h>

#define OUT_HW 56
#define S_  2
#define N_  8
#define C_  32
#define M_  64
#define HF  128
#define WF  128

typedef unsigned int v4u __attribute__((ext_vector_type(4)));
typedef int          v8i __attribute__((ext_vector_type(8)));
typedef int          v4i __attribute__((ext_vector_type(4)));
typedef float        v4f __attribute__((ext_vector_type(4)));

struct alignas(16) AxisE { int i0; int i1; float w; float pad; };

__global__ __launch_bounds__(256) void ROIResize_60464549593192_kernel(
    const float* __restrict__ fmap, const int* __restrict__ boxes,
    float* __restrict__ out)
{
    __shared__ float tile[31 * 31 + 15];      // packed crop tile (<= 961 f32)
    __shared__ AxisE ay[OUT_HW];              // per-oh {y0, y1, wy}
    __shared__ AxisE ax[OUT_HW];              // per-ow {x0, x1, wx}

    const int tid = threadIdx.x;
    const int b   = blockIdx.x;                  // [0, 32768)
    const int nm  = (b >> 5) & (N_ * M_ - 1);
    const int s   = b >> 14;
    const int n   = nm >> 6;
    const int m   = nm & (M_ - 1);
    const int c   = b & (C_ - 1);

    // Box is block-uniform; force into SGPRs so the TDM descriptor is scalar.
    const int* bx = boxes + ((((s * N_ + n) * M_) + m) << 2);
    const int x1b = __builtin_amdgcn_readfirstlane(bx[0]);
    const int y1b = __builtin_amdgcn_readfirstlane(bx[1]);
    const int x2b = __builtin_amdgcn_readfirstlane(bx[2]);
    const int y2b = __builtin_amdgcn_readfirstlane(bx[3]);
    const int cw  = x2b - x1b;                   // crop width  [8,32)
    const int chh = y2b - y1b;                   // crop height [8,32)

    const float* src = fmap
        + (size_t)((s * N_ + n) * C_ + c) * (HF * WF)
        + (size_t)y1b * WF + x1b;                // crop origin

    // PyTorch bilinear (align_corners=False) per-axis tables, crop-relative.
    if (tid < OUT_HW) {
        AxisE e;
        float sy = (float)chh * (1.0f / 56.0f);
        float fy = fmaxf(((float)tid + 0.5f) * sy - 0.5f, 0.0f);
        e.i0 = min((int)fy, chh - 1);
        e.i1 = min(e.i0 + 1, chh - 1);
        e.w  = fy - (float)e.i0;
        e.pad = 0.0f;
        ay[tid] = e;                             // one ds_store_b128

        float sx = (float)cw * (1.0f / 56.0f);
        float fx = fmaxf(((float)tid + 0.5f) * sx - 0.5f, 0.0f);
        e.i0 = min((int)fx, cw - 1);
        e.i1 = min(e.i0 + 1, cw - 1);
        e.w  = fx - (float)e.i0;
        e.pad = 0.0f;
        ax[tid] = e;                             // one ds_store_b128
    }

#if __has_builtin(__builtin_amdgcn_tensor_load_to_lds)
    // ---- CDNA5 Tensor Data Mover: DMA the cw x chh f32 crop tile into LDS ----
    if (tid < 32) {                              // wave 0 issues the TDM op
        unsigned long long ga = (unsigned long long)(uintptr_t)src;
        unsigned lds = (unsigned)(uintptr_t)&tile[0];   // low 32 bits = LDS offset

        // D# group 0: count=1 | lds_addr | global_addr[56:0] | type=2 ("image")
        v4u g0 = { 1u,
                   lds,
                   (unsigned)(ga & 0xFFFFFFFFull),
                   (unsigned)((ga >> 32) & 0x01FFFFFFull) | (2u << 30) };

        // D# group 1: mask=0, data_size=4B, dims/strides of a 2-D tensor tile
        v8i g1;
        g1[0] = (int)(2u << 16);                         // data_size = 2 -> 4B
        g1[1] = (int)((unsigned)WF << 16);               // tensor_dim0 lo16
        g1[2] = (int)((unsigned)HF << 16);               // dim0 hi16=0 | tensor_dim1 lo16
        g1[3] = (int)((unsigned)cw << 16);               // dim1 hi16=0 | tile_dim0 = cw
        g1[4] = (int)(unsigned)chh;                      // tile_dim1 = chh, tile_dim2 = 0
        g1[5] = WF;                                      // tensor_dim0_stride = 128
        g1[6] = (int)(((unsigned)(HF * WF) & 0xFFFFu) << 16); // stride0 hi16=0 | stride1 lo16
        g1[7] = (int)((unsigned)(HF * WF) >> 16);        // stride1 hi
        v4i g2 = { 0, 0, 0, 0 };                         // unused dims (2-D tensor)
        v4i g3 = { 0, 0, 0, 0 };
        v8i g4 = { 0, 0, 0, 0, 0, 0, 0, 0 };             // extra group (clang-23 6-arg form)
        __builtin_amdgcn_tensor_load_to_lds(g0, g1, g2, g3, g4, 0);
        __builtin_amdgcn_s_wait_tensorcnt(0);
    }
#else
    // Fallback: cooperative staged load (no TDM builtin on this toolchain)
    for (int i = tid; i < cw * chh; i += 256) {
        int yy = i / cw;
        tile[i] = src[yy * WF + (i - yy * cw)];
    }
#endif
    __syncthreads();

    // b already equals ((s*512 + nm)*32 + c): output plane is contiguous.
    float* ob = out + (size_t)b * (OUT_HW * OUT_HW);

    // 56x56 plane as 784 chunks of 4 contiguous pixels (each chunk within one
    // output row since 56 % 4 == 0) -> one b128 NT store per chunk per lane.
    for (int chunk = tid; chunk < (OUT_HW * OUT_HW / 4); chunk += 256) {
        int oh  = chunk / (OUT_HW / 4);
        int owb = (chunk - oh * (OUT_HW / 4)) << 2;

        AxisE ye = ay[oh];                       // one ds_load_b128
        const float* r0 = &tile[ye.i0 * cw];
        const float* r1 = &tile[ye.i1 * cw];
        const float  fy = ye.w;

        v4f res;
        #pragma unroll
        for (int j = 0; j < 4; ++j) {
            AxisE xe = ax[owb + j];              // one ds_load_b128
            float v00 = r0[xe.i0], v01 = r0[xe.i1];
            float v10 = r1[xe.i0], v11 = r1[xe.i1];
            float lft = v00 + fy * (v10 - v00);  // y-lerp (matches reference order)
            float rgt = v01 + fy * (v11 - v01);
            res[j] = lft + xe.w * (rgt - lft);   // x-lerp
        }

        // Output is write-once and > L2: 16B non-temporal store keeps L2 for fmap.
        __builtin_nontemporal_store(res, (v4f*)(ob + oh * OUT_HW + owb));
    }
}

extern "C" void kernel_launch(void* const* d_in, const int* in_sizes, int n_in,
                              void* d_out, int out_size, void* d_ws, size_t ws_size,
                              hipStream_t stream) {
    (void)in_sizes; (void)n_in; (void)out_size; (void)d_ws; (void)ws_size;
    const float* fmap  = (const float*)d_in[0];
    const int*   boxes = (const int*)d_in[1];
    float*       out   = (float*)d_out;
    const int n_blocks = S_ * N_ * M_ * C_;      // 32768
    ROIResize_60464549593192_kernel<<<n_blocks, 256, 0, stream>>>(fmap, boxes, out);
}